// AdaptiveEdgeSparsifier_7499012899230
// MI455X (gfx1250) — compile-verified
//
#include <hip/hip_runtime.h>
#include <stdint.h>

#define N 4096
#define THREADS 256
#define KSEL 2867  // max(1, int(4096 * (1 - 0.3)))

#define GLOBAL_AS __attribute__((address_space(1)))
#define LDS_AS    __attribute__((address_space(3)))

typedef int v4i __attribute__((ext_vector_type(4)));

#if defined(__has_builtin)
#  if __has_builtin(__builtin_amdgcn_global_load_async_to_lds_b128)
#    define USE_ASYNC_LDS 1
#  else
#    define USE_ASYNC_LDS 0
#  endif
#  if __has_builtin(__builtin_amdgcn_s_wait_asynccnt)
#    define HAVE_WAIT_ASYNCCNT_BUILTIN 1
#  else
#    define HAVE_WAIT_ASYNCCNT_BUILTIN 0
#  endif
#else
#  define USE_ASYNC_LDS 0
#  define HAVE_WAIT_ASYNCCNT_BUILTIN 0
#endif

// Monotone bijection from float bits to uint32 so that unsigned compare
// of keys == float compare of values (ascending).
__device__ __forceinline__ uint32_t float_key(float x) {
  uint32_t u = __float_as_uint(x);
  return (u & 0x80000000u) ? ~u : (u | 0x80000000u);
}

__global__ __launch_bounds__(THREADS) void edge_sparsify_kernel(
    const float* __restrict__ in, float* __restrict__ out) {
  __shared__ float row[N];          // 16 KB: the row, staged once from HBM
  __shared__ uint32_t hist[256];    // radix histogram
  __shared__ uint32_t scan[256];    // suffix-scan workspace
  __shared__ uint32_t s_prefix;
  __shared__ uint32_t s_remaining;

  const uint32_t tid = threadIdx.x;
  const uint64_t base = (uint64_t)blockIdx.x * (uint64_t)N;
  const float* gin = in + base;
  float* gout = out + base;

  // ---- Stage 1: row -> LDS (async global->LDS, tracked by ASYNCcnt) ----
#if USE_ASYNC_LDS
  {
#pragma unroll
    for (int c = 0; c < 4; ++c) {
      const uint32_t e = (uint32_t)c * 1024u + tid * 4u;  // coalesced float4
      GLOBAL_AS v4i* g = (GLOBAL_AS v4i*)(uintptr_t)(gin + e);
      LDS_AS v4i* l = (LDS_AS v4i*)(uint32_t)(uintptr_t)(&row[e]);
      __builtin_amdgcn_global_load_async_to_lds_b128(g, l, 0, 0);
    }
#if HAVE_WAIT_ASYNCCNT_BUILTIN
    __builtin_amdgcn_s_wait_asynccnt(0);
#else
    asm volatile("s_wait_asynccnt 0" ::: "memory");
#endif
  }
#else
#pragma unroll
  for (int c = 0; c < 4; ++c) {
    const uint32_t e = (uint32_t)c * 1024u + tid * 4u;
    const float4 v = *(const float4*)(gin + e);
    *(float4*)(&row[e]) = v;
  }
#endif
  __syncthreads();

  // ---- Stage 2: MSB-first 4x8-bit radix select for the KSEL-th largest key ----
  uint32_t prefix = 0;
  uint32_t remaining = KSEL;
  for (int pass = 0; pass < 4; ++pass) {
    const int shift = 24 - 8 * pass;
    hist[tid] = 0u;
    __syncthreads();
#pragma unroll
    for (int c = 0; c < 4; ++c) {
      const uint32_t e = (uint32_t)c * 1024u + tid * 4u;
      const float4 v = *(const float4*)(&row[e]);
      const float xv[4] = {v.x, v.y, v.z, v.w};
#pragma unroll
      for (int j = 0; j < 4; ++j) {
        const uint32_t kk = float_key(xv[j]);
        // 64-bit shift: well-defined for shift+8 == 32 on pass 0
        const uint32_t upper = (uint32_t)(((uint64_t)kk) >> (shift + 8));
        if (upper == prefix) atomicAdd(&hist[(kk >> shift) & 0xFFu], 1u);
      }
    }
    __syncthreads();

    // Suffix (from-the-top) inclusive scan: scan[b] = sum of hist[b..255]
    scan[tid] = hist[tid];
    __syncthreads();
    for (int off = 1; off < 256; off <<= 1) {
      const uint32_t v0 = scan[tid];
      const uint32_t v1 = (tid + (uint32_t)off < 256u) ? scan[tid + off] : 0u;
      __syncthreads();
      scan[tid] = v0 + v1;
      __syncthreads();
    }

    // Exactly one bucket satisfies: count(> bucket) < remaining <= count(>= bucket)
    const uint32_t cnt_ge = scan[tid];
    const uint32_t cnt_gt = (tid == 255u) ? 0u : scan[tid + 1];
    if (cnt_ge >= remaining && cnt_gt < remaining) {
      s_prefix = (prefix << 8) | tid;
      s_remaining = remaining - cnt_gt;
    }
    __syncthreads();
    prefix = s_prefix;
    remaining = s_remaining;
  }

  const uint32_t kth_key = prefix;  // key of the k-th largest element

  // ---- Stage 3: threshold from LDS, write out (coalesced float4) ----
#pragma unroll
  for (int c = 0; c < 4; ++c) {
    const uint32_t e = (uint32_t)c * 1024u + tid * 4u;
    const float4 v = *(const float4*)(&row[e]);
    float4 o;
    o.x = (float_key(v.x) >= kth_key) ? v.x : 0.0f;
    o.y = (float_key(v.y) >= kth_key) ? v.y : 0.0f;
    o.z = (float_key(v.z) >= kth_key) ? v.z : 0.0f;
    o.w = (float_key(v.w) >= kth_key) ? v.w : 0.0f;
    *(float4*)(gout + e) = o;
  }
}

extern "C" void kernel_launch(void* const* d_in, const int* in_sizes, int n_in,
                              void* d_out, int out_size, void* d_ws, size_t ws_size,
                              hipStream_t stream) {
  const float* adj = (const float*)d_in[0];
  float* out = (float*)d_out;
  const int rows = in_sizes[0] / N;  // 8 * 4096 = 32768 rows
  edge_sparsify_kernel<<<rows, THREADS, 0, stream>>>(adj, out);
}